// Temporal_FC1_GAP_89833535963819
// MI455X (gfx1250) — compile-verified
//
#include <hip/hip_runtime.h>
#include <hip/hip_bf16.h>

// Problem constants (match reference)
#define BB 32
#define TT 2048
#define CC 1152
#define OO 29
#define TCHUNKS 32
#define TPER (TT / TCHUNKS)   // 64 timesteps per chunk

typedef float v2f __attribute__((ext_vector_type(2)));
typedef float v8f __attribute__((ext_vector_type(8)));

// ---------------------------------------------------------------------------
// Stage 1: partial reduction over T.  part[chunk][b][c] = sum over its T-slice.
// Lanes span c (float4 => 512B contiguous per wave instruction, fully coalesced).
// grid.x = (B*C/4)/256 = 36, grid.y = TCHUNKS = 32  -> 1152 blocks, 294912 thr.
// ---------------------------------------------------------------------------
__global__ __launch_bounds__(256) void reduce_t_kernel(
    const float* __restrict__ x, float* __restrict__ part) {
  const int col4 = blockIdx.x * blockDim.x + threadIdx.x;   // 0 .. B*C/4-1
  const int chunk = blockIdx.y;
  const int C4 = CC / 4;
  const int b  = col4 / C4;
  const int c4 = col4 % C4;

  const float4* p = (const float4*)(x + (size_t)b * TT * CC) + c4
                    + (size_t)(chunk * TPER) * C4;
  float4 s; s.x = 0.f; s.y = 0.f; s.z = 0.f; s.w = 0.f;
#pragma unroll 4
  for (int t = 0; t < TPER; ++t) {
    float4 v = p[(size_t)t * C4];
    s.x += v.x; s.y += v.y; s.z += v.z; s.w += v.w;
  }
  ((float4*)(part + (size_t)chunk * BB * CC))[col4] = s;
}

// ---------------------------------------------------------------------------
// Stage 2: fold partials (deterministic order) and apply affine:
//   A[b,c] = (sum_t x[b,t,c]) / 64 - 2*T
// ---------------------------------------------------------------------------
__global__ __launch_bounds__(256) void finalize_kernel(
    const float* __restrict__ part, float* __restrict__ Amat) {
  const int i = blockIdx.x * blockDim.x + threadIdx.x;      // 0 .. B*C-1
  if (i >= BB * CC) return;
  float s = 0.f;
#pragma unroll
  for (int ch = 0; ch < TCHUNKS; ++ch)
    s += part[(size_t)ch * BB * CC + i];
  Amat[i] = s * (1.0f / 64.0f) - 2.0f * (float)TT;
}

// ---------------------------------------------------------------------------
// Stage 3: z[b,o] = sum_c A[b,c] * W[o,c]  via V_WMMA_F32_16X16X4_F32,
// then epilogue z = (z + T*bias[o]) / length[b].
// One block, 4 waves; wave (mt,nt) owns the 16x16 tile (mt*16, nt*16).
// O=29 padded to 32 with zeroed B columns (multiply-select, no EXEC change:
// WMMA requires EXEC all-ones).
// ---------------------------------------------------------------------------
__global__ __launch_bounds__(128) void gemm_wmma_kernel(
    const float* __restrict__ Amat, const float* __restrict__ W,
    const float* __restrict__ bias, const int* __restrict__ length,
    float* __restrict__ out) {
  const int lane   = threadIdx.x & 31;
  const int wave   = threadIdx.x >> 5;       // 0..3
  const int mtile  = (wave >> 1) * 16;       // 0 or 16
  const int ntile  = (wave & 1) * 16;        // 0 or 16
  const int half   = lane >> 4;              // 0: K 0/1, 1: K 2/3
  const int lane16 = lane & 15;

  const int m = mtile + lane16;              // A row held by this lane
  const int n = ntile + lane16;              // B col held by this lane
  const int koff = half * 2;

  const float* Arow = Amat + (size_t)m * CC;
  const int   nc    = (n < OO) ? n : (OO - 1);
  const float nsc   = (n < OO) ? 1.0f : 0.0f;   // zero-pad cols o>=29
  const float* Wrow = W + (size_t)nc * CC;

  v8f acc = {};
  for (int k = 0; k < CC; k += 4) {
    v2f a, bf;
    a.x  = Arow[k + koff];
    a.y  = Arow[k + koff + 1];
    bf.x = Wrow[k + koff] * nsc;
    bf.y = Wrow[k + koff + 1] * nsc;
    // D = A(16x4,f32) * B(4x16,f32) + C(16x16,f32)
    acc = __builtin_amdgcn_wmma_f32_16x16x4_f32(
        /*neg_a=*/false, a, /*neg_b=*/false, bf,
        /*c_mod=*/(short)0, acc, /*reuse_a=*/false, /*reuse_b=*/false);
  }

  // Epilogue: acc[r] = z[mtile + r + half*8][n]
  const float Tf = (float)TT;
  const float bn = (n < OO) ? bias[n] : 0.0f;
#pragma unroll
  for (int r = 0; r < 8; ++r) {
    const int row = mtile + r + half * 8;    // 0..31
    const float len = (float)length[row];
    const float z = (acc[r] + Tf * bn) / len;
    if (n < OO) out[row * OO + n] = z;       // post-WMMA divergence is fine
  }
}

// ---------------------------------------------------------------------------
extern "C" void kernel_launch(void* const* d_in, const int* in_sizes, int n_in,
                              void* d_out, int out_size, void* d_ws, size_t ws_size,
                              hipStream_t stream) {
  const float* x      = (const float*)d_in[0];   // [B,T,C] f32
  const int*   length = (const int*)  d_in[1];   // [B] int
  const float* W      = (const float*)d_in[2];   // [O,C] f32
  const float* bias   = (const float*)d_in[3];   // [O]   f32
  float* out = (float*)d_out;                    // [B,O] f32

  float* part = (float*)d_ws;                              // [TCHUNKS][B][C]
  float* Amat = part + (size_t)TCHUNKS * BB * CC;          // [B][C]

  dim3 g1((BB * CC / 4) / 256, TCHUNKS);                   // 36 x 32 blocks
  reduce_t_kernel<<<g1, 256, 0, stream>>>(x, part);
  finalize_kernel<<<(BB * CC + 255) / 256, 256, 0, stream>>>(part, Amat);
  gemm_wmma_kernel<<<1, 128, 0, stream>>>(Amat, W, bias, length, out);
}